// EVN_ResnetPointnet_51556787421346
// MI455X (gfx1250) — compile-verified
//
#include <hip/hip_runtime.h>
#include <hip/hip_bf16.h>
#include <math.h>

// ---------------------------------------------------------------------------
// Problem constants (from reference): B=8, N=1024, K=20, H=128
// Token layouts:
//   vector tokens t = (b*3+d)*1024 + n   -> Ttok = 24576
//   norm   tokens u = b*1024 + n         -> U    = 8192
// Activations stored row-major [C][Ttok]; norms/q as [C][U].
// ---------------------------------------------------------------------------
#define BB    8
#define NN    1024
#define KNN   20
#define TTOK  24576
#define UU    8192
#define EPSV  1e-8f
#define SLOPE 0.2f

typedef float v2f __attribute__((ext_vector_type(2)));
typedef float v8f __attribute__((ext_vector_type(8)));

__device__ __forceinline__ v8f wmma_f32_4(v2f a, v2f b, v8f c) {
  // V_WMMA_F32_16X16X4_F32 : D(16x16 f32) = A(16x4 f32) * B(4x16 f32) + C
  return __builtin_amdgcn_wmma_f32_16x16x4_f32(false, a, false, b, (short)0, c,
                                               false, false);
}

// ---------------------------------------------------------------------------
// 1) Brute-force KNN (K=20 smallest squared distances, self included)
// ---------------------------------------------------------------------------
__global__ void evn_knn_kernel(const float* __restrict__ p, int* __restrict__ idx) {
  const int b = blockIdx.x / (NN / 256);
  const int n = (blockIdx.x % (NN / 256)) * 256 + threadIdx.x;
  const float* pb = p + (size_t)b * NN * 3;
  const float qx = pb[n * 3 + 0];
  const float qy = pb[n * 3 + 1];
  const float qz = pb[n * 3 + 2];

  float bestd[KNN];
  int   besti[KNN];
#pragma unroll
  for (int i = 0; i < KNN; ++i) { bestd[i] = 3.0e38f; besti[i] = 0; }

  __shared__ float sx[256], sy[256], sz[256];
  for (int tile = 0; tile < NN; tile += 256) {
    const int m = tile + threadIdx.x;
    sx[threadIdx.x] = pb[m * 3 + 0];
    sy[threadIdx.x] = pb[m * 3 + 1];
    sz[threadIdx.x] = pb[m * 3 + 2];
    __syncthreads();
    for (int j = 0; j < 256; ++j) {
      const float dx = qx - sx[j];
      const float dy = qy - sy[j];
      const float dz = qz - sz[j];
      const float d = dx * dx + dy * dy + dz * dz;
      if (d < bestd[KNN - 1]) {
        int pos = KNN - 1;
        while (pos > 0 && bestd[pos - 1] > d) {
          bestd[pos] = bestd[pos - 1];
          besti[pos] = besti[pos - 1];
          --pos;
        }
        bestd[pos] = d;
        besti[pos] = tile + j;
      }
    }
    __syncthreads();
  }
#pragma unroll
  for (int i = 0; i < KNN; ++i) idx[((size_t)b * NN + n) * KNN + i] = besti[i];
}

// ---------------------------------------------------------------------------
// 2) Fused edge-conv + magnl + mean-over-k.  One wave per (b,n).
//    Low register pressure version: 2 passes of 4 M-tiles (32 acc VGPRs),
//    all rolled loops pinned with unroll(1) so the unroller cannot inflate
//    live ranges; epilogue immediately after each pass ends acc live ranges.
// ---------------------------------------------------------------------------
__global__ void evn_conv_kernel(const float* __restrict__ p,
                                const int*   __restrict__ idx,
                                const float* __restrict__ Wc,   // [128][2]
                                const float* __restrict__ Wm,   // [128][128]
                                const float* __restrict__ bm,   // [128]
                                float* __restrict__ X0) {       // [128][TTOK]
  const int bn = blockIdx.x;
  const int b = bn >> 10;
  const int n = bn & (NN - 1);
  const int l = threadIdx.x;               // 0..31
  const bool valid = (l < KNN);

  __shared__ float Nm[128 * 32];
  __shared__ float Fm[128 * 32];

  const float* pb = p + (size_t)b * NN * 3;
  const float cx = pb[n * 3 + 0];
  const float cy = pb[n * 3 + 1];
  const float cz = pb[n * 3 + 2];
  float ex = 0.f, ey = 0.f, ez = 0.f;      // nbr - center
  if (valid) {
    const int j = idx[((size_t)b * NN + n) * KNN + l];
    ex = pb[j * 3 + 0] - cx;
    ey = pb[j * 3 + 1] - cy;
    ez = pb[j * 3 + 2] - cz;
  }

#pragma unroll 1
  for (int o = 0; o < 128; ++o) {
    const float w0 = Wc[o * 2 + 0];
    const float w1 = Wc[o * 2 + 1];
    const float x0 = w0 * ex + w1 * cx;
    const float x1 = w0 * ey + w1 * cy;
    const float x2 = w0 * ez + w1 * cz;
    Nm[o * 32 + l] = sqrtf(x0 * x0 + x1 * x1 + x2 * x2 + EPSV);
  }
  __syncthreads();

  const int half = l >> 4;
  const int lm = l & 15;

#pragma unroll 1
  for (int nt = 0; nt < 2; ++nt) {
    const int col = nt * 16 + lm;
#pragma unroll 1
    for (int mh = 0; mh < 2; ++mh) {       // M-tiles mh*4 .. mh*4+3
      v8f acc0 = {}, acc1 = {}, acc2 = {}, acc3 = {};
      const float* wp = Wm + (size_t)(mh * 64 + lm) * 128 + 2 * half;
#pragma unroll 1
      for (int k = 0; k < 128; k += 4) {
        v2f bv;
        bv[0] = Nm[(k + 2 * half + 0) * 32 + col];
        bv[1] = Nm[(k + 2 * half + 1) * 32 + col];
        v2f a0, a1, a2, a3;
        a0[0] = wp[0 * 16 * 128 + k]; a0[1] = wp[0 * 16 * 128 + k + 1];
        a1[0] = wp[1 * 16 * 128 + k]; a1[1] = wp[1 * 16 * 128 + k + 1];
        a2[0] = wp[2 * 16 * 128 + k]; a2[1] = wp[2 * 16 * 128 + k + 1];
        a3[0] = wp[3 * 16 * 128 + k]; a3[1] = wp[3 * 16 * 128 + k + 1];
        acc0 = wmma_f32_4(a0, bv, acc0);
        acc1 = wmma_f32_4(a1, bv, acc1);
        acc2 = wmma_f32_4(a2, bv, acc2);
        acc3 = wmma_f32_4(a3, bv, acc3);
      }
      v8f* accs[4] = {&acc0, &acc1, &acc2, &acc3};
#pragma unroll
      for (int q4 = 0; q4 < 4; ++q4) {
        const v8f acc = *accs[q4];
#pragma unroll
        for (int v = 0; v < 8; ++v) {
          const int row = mh * 64 + q4 * 16 + v + 8 * half;
          const float qv = acc[v] + bm[row];
          const float lr = qv > 0.f ? qv : SLOPE * qv;
          Fm[row * 32 + col] = lr / Nm[row * 32 + col];
        }
      }
    }
  }
  __syncthreads();

  const float inv = 1.0f / (float)KNN;
#pragma unroll 1
  for (int o = 0; o < 128; ++o) {
    const float w0 = Wc[o * 2 + 0];
    const float w1 = Wc[o * 2 + 1];
    const float f = valid ? Fm[o * 32 + l] : 0.f;
    float s0 = (w0 * ex + w1 * cx) * f;
    float s1 = (w0 * ey + w1 * cy) * f;
    float s2 = (w0 * ez + w1 * cz) * f;
#pragma unroll
    for (int off = 16; off > 0; off >>= 1) {
      s0 += __shfl_xor(s0, off, 32);
      s1 += __shfl_xor(s1, off, 32);
      s2 += __shfl_xor(s2, off, 32);
    }
    if (l == 0) {
      X0[(size_t)o * TTOK + (b * 3 + 0) * NN + n] = s0 * inv;
      X0[(size_t)o * TTOK + (b * 3 + 1) * NN + n] = s1 * inv;
      X0[(size_t)o * TTOK + (b * 3 + 2) * NN + n] = s2 * inv;
    }
  }
}

// ---------------------------------------------------------------------------
// 3) Generic fp32 WMMA GEMM:  C[M][T] (+= / =) A[M][Kd] * B[Kd][T] (+ bias[M])
//    grid = (ceil(T/64), M/16), block = 32 (one wave), 16x64 tile,
//    2-stage software pipeline (unroll-capped), hoisted/clamped col guards.
// ---------------------------------------------------------------------------
__global__ void evn_gemm_wmma_kernel(const float* __restrict__ A,
                                     const float* __restrict__ Bm,
                                     float* __restrict__ C,
                                     const float* __restrict__ bias,
                                     int Kd, int T, int addC) {
  const int l = threadIdx.x;
  const int half = l >> 4;
  const int lm = l & 15;
  const int m0 = blockIdx.y * 16;
  const int t0 = blockIdx.x * 64;

  int tc0, tc1, tc2, tc3;
  {
    int t;
    t = t0 + 0 * 16 + lm; tc0 = (t < T) ? t : 0;
    t = t0 + 1 * 16 + lm; tc1 = (t < T) ? t : 0;
    t = t0 + 2 * 16 + lm; tc2 = (t < T) ? t : 0;
    t = t0 + 3 * 16 + lm; tc3 = (t < T) ? t : 0;
  }
  const float* ap = A + (size_t)(m0 + lm) * Kd + 2 * half;

  v8f acc0 = {}, acc1 = {}, acc2 = {}, acc3 = {};

  // stage-0 loads (k = 0)
  v2f a, b0, b1, b2, b3;
  a[0] = ap[0];
  a[1] = ap[1];
  {
    const size_t r0 = (size_t)(2 * half) * T;
    const size_t r1 = r0 + T;
    b0[0] = Bm[r0 + tc0]; b0[1] = Bm[r1 + tc0];
    b1[0] = Bm[r0 + tc1]; b1[1] = Bm[r1 + tc1];
    b2[0] = Bm[r0 + tc2]; b2[1] = Bm[r1 + tc2];
    b3[0] = Bm[r0 + tc3]; b3[1] = Bm[r1 + tc3];
  }

#pragma unroll 1
  for (int k = 0; k < Kd - 4; k += 4) {
    // prefetch k+4
    v2f an, c0, c1, c2, c3;
    an[0] = ap[k + 4];
    an[1] = ap[k + 5];
    const size_t r0 = (size_t)(k + 4 + 2 * half) * T;
    const size_t r1 = r0 + T;
    c0[0] = Bm[r0 + tc0]; c0[1] = Bm[r1 + tc0];
    c1[0] = Bm[r0 + tc1]; c1[1] = Bm[r1 + tc1];
    c2[0] = Bm[r0 + tc2]; c2[1] = Bm[r1 + tc2];
    c3[0] = Bm[r0 + tc3]; c3[1] = Bm[r1 + tc3];
    // compute k
    acc0 = wmma_f32_4(a, b0, acc0);
    acc1 = wmma_f32_4(a, b1, acc1);
    acc2 = wmma_f32_4(a, b2, acc2);
    acc3 = wmma_f32_4(a, b3, acc3);
    a = an; b0 = c0; b1 = c1; b2 = c2; b3 = c3;
  }
  acc0 = wmma_f32_4(a, b0, acc0);
  acc1 = wmma_f32_4(a, b1, acc1);
  acc2 = wmma_f32_4(a, b2, acc2);
  acc3 = wmma_f32_4(a, b3, acc3);

  v8f* accs[4] = {&acc0, &acc1, &acc2, &acc3};
#pragma unroll
  for (int nt = 0; nt < 4; ++nt) {
    const int t = t0 + nt * 16 + lm;
    if (t >= T) continue;
    const v8f acc = *accs[nt];
#pragma unroll
    for (int v = 0; v < 8; ++v) {
      const int row = m0 + v + 8 * half;
      float val = acc[v];
      if (bias) val += bias[row];
      if (addC) val += C[(size_t)row * T + t];
      C[(size_t)row * T + t] = val;
    }
  }
}

// ---------------------------------------------------------------------------
// 4) Elementwise / reduction helpers
// ---------------------------------------------------------------------------
__global__ void evn_norm_kernel(const float* __restrict__ X,
                                float* __restrict__ Nr, int Cn) {
  const int i = blockIdx.x * 256 + threadIdx.x;
  if (i >= Cn * UU) return;
  const int c = i / UU, u = i % UU;
  const int b = u >> 10, n = u & (NN - 1);
  const float* xr = X + (size_t)c * TTOK + (size_t)b * 3 * NN + n;
  const float x0 = xr[0], x1 = xr[NN], x2 = xr[2 * NN];
  Nr[(size_t)c * UU + u] = sqrtf(x0 * x0 + x1 * x1 + x2 * x2 + EPSV);
}

__global__ void evn_magfin_kernel(const float* __restrict__ X,
                                  const float* __restrict__ Nr,
                                  const float* __restrict__ Q,
                                  float* __restrict__ Y, int Cn) {
  const int i = blockIdx.x * 256 + threadIdx.x;
  if (i >= Cn * TTOK) return;
  const int c = i / TTOK, t = i % TTOK;
  const int b = t / (3 * NN);
  const int n = t & (NN - 1);
  const int u = b * NN + n;
  const float nv = Nr[(size_t)c * UU + u];
  const float q = Q[(size_t)c * UU + u];
  const float lr = q > 0.f ? q : SLOPE * q;
  Y[i] = X[i] / nv * lr;
}

__global__ void evn_poolcat_kernel(float* __restrict__ Y) {
  const int cbd = blockIdx.x;          // c*24 + (b*3+d)
  const int c = cbd / 24, bd = cbd % 24;
  const float* src = Y + (size_t)c * TTOK + (size_t)bd * NN;
  float s = 0.f;
  for (int n = threadIdx.x; n < NN; n += 128) s += src[n];
  __shared__ float red[128];
  red[threadIdx.x] = s;
  __syncthreads();
  for (int off = 64; off > 0; off >>= 1) {
    if (threadIdx.x < off) red[threadIdx.x] += red[threadIdx.x + off];
    __syncthreads();
  }
  const float mean = red[0] * (1.0f / (float)NN);
  float* dst = Y + (size_t)(128 + c) * TTOK + (size_t)bd * NN;
  for (int n = threadIdx.x; n < NN; n += 128) dst[n] = mean;
}

__global__ void evn_meanN_kernel(const float* __restrict__ Y,
                                 float* __restrict__ Mo) {
  const int cbd = blockIdx.x;
  const int c = cbd / 24, bd = cbd % 24;
  const float* src = Y + (size_t)c * TTOK + (size_t)bd * NN;
  float s = 0.f;
  for (int n = threadIdx.x; n < NN; n += 128) s += src[n];
  __shared__ float red[128];
  red[threadIdx.x] = s;
  __syncthreads();
  for (int off = 64; off > 0; off >>= 1) {
    if (threadIdx.x < off) red[threadIdx.x] += red[threadIdx.x + off];
    __syncthreads();
  }
  if (threadIdx.x == 0) Mo[c * 24 + bd] = red[0] * (1.0f / (float)NN);
}

__global__ void evn_ncalc_kernel(const float* __restrict__ Mo,
                                 float* __restrict__ nc) {
  const int i = blockIdx.x * 256 + threadIdx.x;
  if (i >= 128 * 8) return;
  const int c = i >> 3, b = i & 7;
  const float* m = Mo + c * 24 + b * 3;
  nc[c * 8 + b] = sqrtf(m[0] * m[0] + m[1] * m[1] + m[2] * m[2] + EPSV);
}

__global__ void evn_yfin_kernel(const float* __restrict__ Mo,
                                const float* __restrict__ nc,
                                const float* __restrict__ qc,
                                float* __restrict__ y) {
  const int i = blockIdx.x * 256 + threadIdx.x;
  if (i >= 128 * 24) return;
  const int c = i / 24, bd = i % 24, b = bd / 3;
  const float nv = nc[c * 8 + b];
  const float q = qc[c * 8 + b];
  const float lr = q > 0.f ? q : SLOPE * q;
  y[i] = Mo[i] / nv * lr;
}

__global__ void evn_writeout_kernel(const float* __restrict__ Of,
                                    float* __restrict__ out) {
  const int i = blockIdx.x * 256 + threadIdx.x;
  if (i >= BB * 128 * 3) return;
  const int b = i / 384;
  const int r = i % 384;
  const int o = r / 3, d = r % 3;
  out[i] = Of[o * 24 + b * 3 + d];
}

// ---------------------------------------------------------------------------
// Host orchestration
// ---------------------------------------------------------------------------
extern "C" void kernel_launch(void* const* d_in, const int* in_sizes, int n_in,
                              void* d_out, int out_size, void* d_ws, size_t ws_size,
                              hipStream_t stream) {
  (void)in_sizes; (void)n_in; (void)out_size; (void)ws_size;
  const float* p       = (const float*)d_in[0];
  const float* W_conv  = (const float*)d_in[1];   // 128x2
  const float* Wm_conv = (const float*)d_in[2];   // 128x128
  const float* bm_conv = (const float*)d_in[3];   // 128
  const float* W_fcpos = (const float*)d_in[4];   // 256x128
  const float* Wfc0    = (const float*)d_in[5];   // 5x128x256
  const float* Wfc1    = (const float*)d_in[6];   // 5x128x128
  const float* Wsc     = (const float*)d_in[7];   // 5x128x256
  const float* Wm0     = (const float*)d_in[8];   // 5x256x256
  const float* bm0     = (const float*)d_in[9];   // 5x256
  const float* Wm1     = (const float*)d_in[10];  // 5x128x128
  const float* bm1     = (const float*)d_in[11];  // 5x128
  const float* Wm_c    = (const float*)d_in[12];  // 128x128
  const float* bm_c    = (const float*)d_in[13];  // 128
  const float* W_fcc   = (const float*)d_in[14];  // 128x128

  int*   idxb = (int*)d_ws;
  float* fw   = (float*)d_ws;
  size_t off = 262144;                               // 1 MB / 4
  float* Abuf = fw + off; off += (size_t)256 * TTOK;
  float* Bbuf = fw + off; off += (size_t)256 * TTOK;
  float* Hbuf = fw + off; off += (size_t)128 * TTOK;
  float* NR   = fw + off; off += (size_t)256 * UU;
  float* Qb   = fw + off; off += (size_t)256 * UU;
  float* Mo   = fw + off; off += 3072;
  float* ncb  = fw + off; off += 1024;
  float* qcb  = fw + off; off += 1024;
  float* ybuf = fw + off; off += 3072;
  float* Ofb  = fw + off; off += 3072;

  auto gemm = [&](const float* Aw, const float* Bx, float* Cy, const float* bias,
                  int M, int Kd, int T, int addC) {
    dim3 g((T + 63) / 64, M / 16);
    evn_gemm_wmma_kernel<<<g, dim3(32), 0, stream>>>(Aw, Bx, Cy, bias, Kd, T, addC);
  };

  // 1) KNN
  evn_knn_kernel<<<dim3(BB * (NN / 256)), dim3(256), 0, stream>>>(p, idxb);

  // 2) edge-conv -> X0 in Bbuf rows 0..127
  evn_conv_kernel<<<dim3(BB * NN), dim3(32), 0, stream>>>(
      p, idxb, W_conv, Wm_conv, bm_conv, Bbuf);

  // 3) fcpos: Abuf(256 rows) = W_fcpos @ X0
  gemm(W_fcpos, Bbuf, Abuf, nullptr, 256, 128, TTOK, 0);

  // 4) residual blocks
  float* cur = Abuf;
  float* nxt = Bbuf;
  for (int i = 0; i < 5; ++i) {
    const float* wm0 = Wm0 + (size_t)i * 256 * 256;
    const float* b0  = bm0 + (size_t)i * 256;
    const float* wf0 = Wfc0 + (size_t)i * 128 * 256;
    const float* wm1 = Wm1 + (size_t)i * 128 * 128;
    const float* b1  = bm1 + (size_t)i * 128;
    const float* wf1 = Wfc1 + (size_t)i * 128 * 128;
    const float* wsc = Wsc + (size_t)i * 128 * 256;

    evn_norm_kernel<<<dim3((256 * UU + 255) / 256), dim3(256), 0, stream>>>(cur, NR, 256);
    gemm(wm0, NR, Qb, b0, 256, 256, UU, 0);
    gemm(wsc, cur, nxt, nullptr, 128, 256, TTOK, 0);
    evn_magfin_kernel<<<dim3((256 * TTOK + 255) / 256), dim3(256), 0, stream>>>(cur, NR, Qb, cur, 256);
    gemm(wf0, cur, Hbuf, nullptr, 128, 256, TTOK, 0);
    evn_norm_kernel<<<dim3((128 * UU + 255) / 256), dim3(256), 0, stream>>>(Hbuf, NR, 128);
    gemm(wm1, NR, Qb, b1, 128, 128, UU, 0);
    evn_magfin_kernel<<<dim3((128 * TTOK + 255) / 256), dim3(256), 0, stream>>>(Hbuf, NR, Qb, Hbuf, 128);
    gemm(wf1, Hbuf, nxt, nullptr, 128, 128, TTOK, 1);
    if (i < 4)
      evn_poolcat_kernel<<<dim3(128 * 24), dim3(128), 0, stream>>>(nxt);
    float* tmp = cur; cur = nxt; nxt = tmp;
  }

  // 5) head
  evn_meanN_kernel<<<dim3(128 * 24), dim3(128), 0, stream>>>(cur, Mo);
  evn_ncalc_kernel<<<dim3(4), dim3(256), 0, stream>>>(Mo, ncb);
  gemm(Wm_c, ncb, qcb, bm_c, 128, 128, 8, 0);
  evn_yfin_kernel<<<dim3(12), dim3(256), 0, stream>>>(Mo, ncb, qcb, ybuf);
  gemm(W_fcc, ybuf, Ofb, nullptr, 128, 128, 24, 0);
  evn_writeout_kernel<<<dim3(12), dim3(256), 0, stream>>>(Ofb, (float*)d_out);
}